// PhysicsLayer_68092411511563
// MI455X (gfx1250) — compile-verified
//
#include <hip/hip_runtime.h>
#include <math.h>

// ---------------------------------------------------------------------------
// Implicit advection-diffusion (backward Euler + Thomas solve), 199 steps.
// Single-WGP persistent kernel; Thomas sweeps parallelized as affine scans.
// Time-constant carry-propagation matrices applied via V_WMMA_F32_16X16X4_F32.
// ---------------------------------------------------------------------------

#define NXC      2048
#define NPTS     2049
#define NTSTEPS  200
#define NSTEPS   199          // NT-1 scan steps in the reference
#define NTHREADS 256
#define SEG      8            // elements per thread
#define PI_F     3.14159265358979323846f

typedef __attribute__((ext_vector_type(2))) float v2f;
typedef __attribute__((ext_vector_type(8))) float v8f;

// workspace layout (float offsets)
#define OFF_A    0            // forward scan coefficient A_i            (2048)
#define OFF_Q    2048         // -invden_i/dt (B_i = u_i * q_i)          (2048)
#define OFF_PF   4096         // in-segment prefix products of A         (2048)
#define OFF_GR   6144         // backward coefficient (reversed index)   (2048)
#define OFF_PB   8192         // in-segment prefix products of G         (2048)
#define OFF_U0   10240        // initial condition                       (2049)
#define OFF_INL  12289        // precomputed inlet values per step       (200)
#define OFF_WF   12489        // per-thread inter-wave weights, fwd      (2048)
#define OFF_WB   14537        // per-thread inter-wave weights, bwd      (2048)
#define OFF_BOP  16586        // WMMA B-operand tables [dir][w][h][c][lane][2] (16384)
#define WS_FLOATS 32970

__device__ __forceinline__ float gaussf(float x, float alpha, float vel, float tt) {
  float four_at = 4.0f * alpha * tt;
  float arg = x - 0.5f - vel * tt;
  return (1.0f / sqrtf(PI_F * four_at)) * expf(-(arg * arg) / four_at);
}

// ---------------------------------------------------------------------------
// Setup: factor the (time-constant) tridiagonal system once, build all scan
// constants, initial condition, inlet series, and WMMA carry operands.
// ---------------------------------------------------------------------------
__global__ __launch_bounds__(NTHREADS) void pl_setup(const float* __restrict__ alpha_p,
                                                     const float* __restrict__ vel_p,
                                                     const float* __restrict__ t_p,
                                                     float* __restrict__ ws) {
  __shared__ float sA[NXC];       // forward A_i
  __shared__ float sCP[NXC];      // cp_i
  __shared__ float sAsegF[NTHREADS];
  __shared__ float sAsegB[NTHREADS];
  __shared__ float sAtF[8];
  __shared__ float sAtB[8];

  const int t = threadIdx.x;
  const float alpha = alpha_p[0], vel = vel_p[0], tt = t_p[0];
  const float dt = (tt - 1.0f) / (float)NTSTEPS;
  const float rd = alpha * (float)NXC * (float)NXC;   // alpha/DX^2
  const float ra = vel * (float)NXC * 0.5f;           // vel/(2 DX)
  const float aco = rd + ra;
  const float bco = -2.0f * rd - 1.0f / dt;
  const float cco = rd - ra;
  const float bN  = -rd - ra;

  if (t == 0) {
    // Forward elimination factors (serial once; row 0: b=1, a=c=0)
    sA[0] = 0.0f; sCP[0] = 0.0f;
    ws[OFF_A + 0] = 0.0f;
    ws[OFF_Q + 0] = -1.0f / dt;   // unused (row 0 RHS is the inlet)
    float cp_prev = 0.0f;
    for (int i = 1; i < NXC; ++i) {
      float bb = (i == NXC - 1) ? bN : bco;
      float cc = (i == NXC - 1) ? 0.0f : cco;
      float den = bb - aco * cp_prev;
      float inv = 1.0f / den;
      float cp = cc * inv;
      float Ai = -aco * inv;
      sA[i] = Ai; sCP[i] = cp;
      ws[OFF_A + i] = Ai;
      ws[OFF_Q + i] = -inv / dt;
      cp_prev = cp;
    }
  }
  // inlet boundary values for every step (hoists expf/sqrtf off the hot loop)
  if (t < NSTEPS) {
    float tn1 = 1.0f + dt * ((float)t + 2.0f);
    ws[OFF_INL + t] = gaussf(0.0f, alpha, vel, tn1);
  }
  // Initial condition u(x, t0=1)
  for (int i = t; i < NPTS; i += NTHREADS) {
    float x = (float)i / (float)NXC;
    ws[OFF_U0 + i] = gaussf(x, alpha, vel, 1.0f);
  }
  __syncthreads();

  // Per-segment prefix products (forward uses A, backward uses reversed -cp)
  {
    float pf = 1.0f, pb = 1.0f;
#pragma unroll
    for (int j = 0; j < SEG; ++j) {
      int i = t * SEG + j;
      pf *= sA[i];
      ws[OFF_PF + i] = pf;
      float g = (i == 0) ? 0.0f : -sCP[NXC - 1 - i];  // reversed index r == i
      ws[OFF_GR + i] = g;
      pb *= g;
      ws[OFF_PB + i] = pb;
    }
    sAsegF[t] = pf;
    sAsegB[t] = pb;
  }
  __syncthreads();

  // Per-wave composite products
  if (t < 8) {
    float p = 1.0f;
    for (int m = 0; m < 32; ++m) p *= sAsegF[t * 32 + m];
    sAtF[t] = p;
  } else if (t < 16) {
    int w = t - 8;
    float p = 1.0f;
    for (int m = 0; m < 32; ++m) p *= sAsegB[w * 32 + m];
    sAtB[w] = p;
  }
  __syncthreads();

  // Per-thread inter-wave weight vectors: carry += sum_k W[k]*Btot[k]
  // W[k] = Qlane * prod_{m=k+1}^{w-1} Atot[m]  for k < w, else 0  (Qlane folded in)
  {
    int w = t >> 5, lam = t & 31;
    float qf = 1.0f, qb = 1.0f;
    for (int m = 0; m < lam; ++m) { qf *= sAsegF[w * 32 + m]; qb *= sAsegB[w * 32 + m]; }
#pragma unroll
    for (int k = 0; k < 8; ++k) {
      float wf = 0.0f, wb = 0.0f;
      if (k < w) {
        float pf = qf, pb = qb;
        for (int m = k + 1; m < w; ++m) { pf *= sAtF[m]; pb *= sAtB[m]; }
        wf = pf; wb = pb;
      }
      ws[OFF_WF + t * 8 + k] = wf;
      ws[OFF_WB + t * 8 + k] = wb;
    }
  }

  // WMMA B-operands: M[n,k] = prod_{m=k+1}^{n-1} Aseg[m] (0 if k>=n), stored in
  // the 4x16 f32 B layout: lanes 0-15 hold K rows {4c,4c+1} at column n=16h+lane,
  // lanes 16-31 hold K rows {4c+2,4c+3} at column n=16h+(lane-16).
  for (int k32 = 0; k32 < 32; ++k32) {
    int e    = t + NTHREADS * k32;     // 0..8191
    int lane = e & 31;
    int c    = (e >> 5) & 7;
    int h    = (e >> 8) & 1;
    int w    = (e >> 9) & 7;
    int dir  = (e >> 12) & 1;
    const float* Aseg = dir ? &sAsegB[w * 32] : &sAsegF[w * 32];
    int n  = 16 * h + (lane & 15);
    int kk = 4 * c + ((lane < 16) ? 0 : 2);
    float v0 = 0.0f, v1 = 0.0f;
    if (kk < n)     { float p = 1.0f; for (int m = kk + 1; m <= n - 1; ++m) p *= Aseg[m]; v0 = p; }
    if (kk + 1 < n) { float p = 1.0f; for (int m = kk + 2; m <= n - 1; ++m) p *= Aseg[m]; v1 = p; }
    ws[OFF_BOP + 2 * e + 0] = v0;
    ws[OFF_BOP + 2 * e + 1] = v1;
  }
}

// ---------------------------------------------------------------------------
// 32-lane affine-scan carry combine via f32 WMMA (split even/odd accumulators
// to halve the RAW chain), plus straight-line inter-wave weighted fixup.
// NOTE: exactly one __syncthreads inside; the sBt-reuse hazard is covered by
// the dp-store / u-store barriers in the caller.
// ---------------------------------------------------------------------------
__device__ __forceinline__ float wave_affine_carry(float Bseg, float Aseg,
                                                   const v2f (&Bop)[2][8],
                                                   const float (&W)[8],
                                                   float* __restrict__ sBt,
                                                   int w, int lam) {
  v8f h0a = {}, h0b = {}, h1a = {}, h1b = {};
#pragma unroll
  for (int c = 0; c < 8; ++c) {
    // A operand: the Bseg vector replicated across rows.
    // lanes 0-15 need v[4c],v[4c+1]; lanes 16-31 need v[4c+2],v[4c+3].
    int base = 4 * c + ((lam < 16) ? 0 : 2);
    v2f a;
    a.x = __shfl(Bseg, base, 32);
    a.y = __shfl(Bseg, base + 1, 32);
    if (c & 1) {
      h0b = __builtin_amdgcn_wmma_f32_16x16x4_f32(false, a, false, Bop[0][c], (short)0, h0b, false, false);
      h1b = __builtin_amdgcn_wmma_f32_16x16x4_f32(false, a, false, Bop[1][c], (short)0, h1b, false, false);
    } else {
      h0a = __builtin_amdgcn_wmma_f32_16x16x4_f32(false, a, false, Bop[0][c], (short)0, h0a, false, false);
      h1a = __builtin_amdgcn_wmma_f32_16x16x4_f32(false, a, false, Bop[1][c], (short)0, h1a, false, false);
    }
  }
  // D VGPR0: lane l holds column (l & 15) of its half -> lane picks its own carry.
  float intraC = (lam < 16) ? (h0a[0] + h0b[0]) : (h1a[0] + h1b[0]);
  if (lam == 31) sBt[w] = fmaf(Aseg, intraC, Bseg);  // wave-local composite B
  __syncthreads();
  float r = intraC;
#pragma unroll
  for (int k = 0; k < 8; ++k) r = fmaf(W[k], sBt[k], r);
  return r;
}

// ---------------------------------------------------------------------------
// Persistent solver: all 199 steps in one workgroup, state in LDS.
// ---------------------------------------------------------------------------
__global__ __launch_bounds__(NTHREADS) void pl_solve(const float* __restrict__ ws,
                                                     float* __restrict__ out) {
  __shared__ float sU[NPTS];
  __shared__ float sDP[NXC];
  __shared__ float sInl[NSTEPS];
  __shared__ float sBt[8];

  const int t = threadIdx.x;
  const int w = t >> 5;
  const int lam = t & 31;

  // time-constant per-thread scan coefficients -> registers
  float Ac[SEG], Qc[SEG], Pf[SEG], Gr[SEG], Pb[SEG], WF[8], WB[8];
#pragma unroll
  for (int j = 0; j < SEG; ++j) {
    int i = t * SEG + j;
    Ac[j] = ws[OFF_A + i];
    Qc[j] = ws[OFF_Q + i];
    Pf[j] = ws[OFF_PF + i];
    Gr[j] = ws[OFF_GR + i];
    Pb[j] = ws[OFF_PB + i];
  }
#pragma unroll
  for (int k = 0; k < 8; ++k) {
    WF[k] = ws[OFF_WF + t * 8 + k];
    WB[k] = ws[OFF_WB + t * 8 + k];
  }

  // constant WMMA B-operands (carry-propagation matrices) -> registers
  v2f BopF[2][8], BopB[2][8];
  const v2f* bop = reinterpret_cast<const v2f*>(ws + OFF_BOP);
#pragma unroll
  for (int h = 0; h < 2; ++h)
#pragma unroll
    for (int c = 0; c < 8; ++c) {
      BopF[h][c] = bop[(((0 * 8 + w) * 2 + h) * 8 + c) * 32 + lam];
      BopB[h][c] = bop[(((1 * 8 + w) * 2 + h) * 8 + c) * 32 + lam];
    }

  for (int i = t; i < NPTS; i += NTHREADS) sU[i] = ws[OFF_U0 + i];
  if (t < NSTEPS) sInl[t] = ws[OFF_INL + t];
  __syncthreads();

  const int i0 = t * SEG;            // forward segment base
  const int rb = NXC - SEG - i0;     // backward segment base (contiguous)
  float4* const pU  = reinterpret_cast<float4*>(sU + i0);
  float4* const pUr = reinterpret_cast<float4*>(sU + rb);
  float4* const pD  = reinterpret_cast<float4*>(sDP + i0);
  const float4* const pDr = reinterpret_cast<const float4*>(sDP + rb);

  float y[SEG];
  for (int n = 0; n < NSTEPS; ++n) {
    float inlet = sInl[n];

    // ---- forward sweep: dp_i = A_i*dp_{i-1} + B_i,  B_i = u_i * q_i ----
    {
      float4 u0v = pU[0], u1v = pU[1];
      float uv[SEG] = {u0v.x, u0v.y, u0v.z, u0v.w, u1v.x, u1v.y, u1v.z, u1v.w};
      y[0] = (t == 0) ? inlet : uv[0] * Qc[0];
#pragma unroll
      for (int j = 1; j < SEG; ++j)
        y[j] = fmaf(Ac[j], y[j - 1], uv[j] * Qc[j]);
    }
    float carry = wave_affine_carry(y[SEG - 1], Pf[SEG - 1], BopF, WF, sBt, w, lam);
    {
      float d[SEG];
#pragma unroll
      for (int j = 0; j < SEG; ++j) d[j] = fmaf(Pf[j], carry, y[j]);
      pD[0] = make_float4(d[0], d[1], d[2], d[3]);
      pD[1] = make_float4(d[4], d[5], d[6], d[7]);
    }
    __syncthreads();

    // ---- backward sweep (reversed index r): x_r = G_r*x_{r-1} + dp[N-1-r] ----
    {
      float4 d0v = pDr[0], d1v = pDr[1];
      // reversed element j lives at linear slot (SEG-1-j) of [d0v,d1v]
      float dr[SEG] = {d1v.w, d1v.z, d1v.y, d1v.x, d0v.w, d0v.z, d0v.y, d0v.x};
      y[0] = dr[0];
#pragma unroll
      for (int j = 1; j < SEG; ++j)
        y[j] = fmaf(Gr[j], y[j - 1], dr[j]);
    }
    float carryb = wave_affine_carry(y[SEG - 1], Pb[SEG - 1], BopB, WB, sBt, w, lam);
    {
      float x[SEG];
#pragma unroll
      for (int j = 0; j < SEG; ++j) x[j] = fmaf(Pb[j], carryb, y[j]);
      pUr[0] = make_float4(x[7], x[6], x[5], x[4]);
      pUr[1] = make_float4(x[3], x[2], x[1], x[0]);
      if (t == 0) sU[NXC] = x[0];    // copy-out BC at right edge
    }
    __syncthreads();
  }

  for (int i = t; i < NPTS; i += NTHREADS) out[i] = sU[i];
}

// ---------------------------------------------------------------------------
extern "C" void kernel_launch(void* const* d_in, const int* in_sizes, int n_in,
                              void* d_out, int out_size, void* d_ws, size_t ws_size,
                              hipStream_t stream) {
  (void)in_sizes; (void)n_in; (void)out_size; (void)ws_size;
  const float* alpha = (const float*)d_in[0];
  const float* vel   = (const float*)d_in[1];
  const float* tP    = (const float*)d_in[2];
  float* ws = (float*)d_ws;
  pl_setup<<<1, NTHREADS, 0, stream>>>(alpha, vel, tP, ws);
  pl_solve<<<1, NTHREADS, 0, stream>>>(ws, (float*)d_out);
}